// GraphDANN_2448131358816
// MI455X (gfx1250) — compile-verified
//
#include <hip/hip_runtime.h>
#include <math.h>

// Problem constants (match reference)
#define NN    20000
#define EE    320000
#define ETOT  (EE + NN)      // with self-loops
#define GG    64
#define HH    4
#define CC    64
#define FDIM  256            // HH*CC
#define SLOPE 0.2f

typedef __attribute__((ext_vector_type(2))) float v2f;
typedef __attribute__((ext_vector_type(8))) float v8f;

// ---------------------------------------------------------------- utilities
__device__ inline float atomicMaxF(float* addr, float value) {
  // Total-order-correct float max via int/uint atomics on IEEE bit patterns.
  if (value >= 0.0f)
    return __int_as_float(atomicMax((int*)addr, __float_as_int(value)));
  return __uint_as_float(atomicMin((unsigned int*)addr, __float_as_uint(value)));
}

__global__ void fill_f32(float* p, float v, int n) {
  int i = blockIdx.x * blockDim.x + threadIdx.x;
  if (i < n) p[i] = v;
}

// ---------------------------------------------------------------- WMMA GEMM
// C[M,Nc] = A[M,K] * B[K,Nc], row-major fp32, using V_WMMA_F32_16X16X4_F32.
// One wave owns a 16-row M tile and sweeps all Nc/16 column tiles.
// A frag (16x4): lane L holds row L&15, K = (L>>4)*2 .. +1 in v[0..1]
// B frag (4x16): lane L holds col L&15, K = (L>>4)*2 .. +1 (column-major frag)
// C/D (16x16):  VGPR j: lanes 0-15 -> M=j, lanes 16-31 -> M=j+8, N=lane&15
__global__ void gemm_wmma_f32(const float* __restrict__ A,
                              const float* __restrict__ B,
                              float* __restrict__ C,
                              int M, int K, int Nc) {
  const int lane  = threadIdx.x & 31;
  const int wave  = threadIdx.x >> 5;
  const int mtile = blockIdx.x * (blockDim.x >> 5) + wave;
  if (mtile * 16 >= M) return;               // wave-uniform guard (EXEC all-1)
  const int m0  = mtile * 16;
  const int r   = lane & 15;
  const int kb  = (lane >> 4) << 1;          // 0 or 2
  const int row = m0 + r;
  const int mhi = (lane >> 4) << 3;          // 0 or 8

  for (int nt = 0; nt < (Nc >> 4); ++nt) {
    const int col = (nt << 4) + r;
    v8f acc = {};
    for (int k = 0; k < K; k += 4) {
      v2f a, b;
      const float* ap = A + (size_t)row * K + k + kb;
      a.x = ap[0];
      a.y = ap[1];
      const float* bp = B + (size_t)(k + kb) * Nc + col;
      b.x = bp[0];
      b.y = bp[Nc];
      acc = __builtin_amdgcn_wmma_f32_16x16x4_f32(false, a, false, b,
                                                  (short)0, acc, false, false);
    }
#pragma unroll
    for (int j = 0; j < 8; ++j)
      C[(size_t)(m0 + j + mhi) * Nc + col] = acc[j];
  }
}

// ---------------------------------------------------------------- GAT pieces
__global__ void node_scores(const float* __restrict__ h,
                            const float* __restrict__ a_src,
                            const float* __restrict__ a_dst,
                            float* __restrict__ es, float* __restrict__ ed) {
  int t = blockIdx.x * blockDim.x + threadIdx.x;
  if (t >= NN * HH) return;
  int n = t >> 2, hd = t & 3;
  const float* hp = h + (size_t)n * FDIM + hd * CC;
  const float* as = a_src + hd * CC;
  const float* ad = a_dst + hd * CC;
  float s = 0.f, d = 0.f;
#pragma unroll 4
  for (int c = 0; c < CC; ++c) { float v = hp[c]; s += v * as[c]; d += v * ad[c]; }
  es[t] = s; ed[t] = d;
}

__device__ inline void edge_nodes(const int* __restrict__ ei, int e, int& s, int& d) {
  if (e < EE) { s = ei[e]; d = ei[EE + e]; }
  else        { s = e - EE; d = s; }          // appended self-loops
}

__global__ void edge_alpha_max(const int* __restrict__ ei,
                               const float* __restrict__ es,
                               const float* __restrict__ ed,
                               float* __restrict__ alpha,
                               float* __restrict__ m) {
  int e = blockIdx.x * blockDim.x + threadIdx.x;
  if (e >= ETOT) return;
  int s, d; edge_nodes(ei, e, s, d);
#pragma unroll
  for (int hd = 0; hd < HH; ++hd) {
    float v = es[s * HH + hd] + ed[d * HH + hd];
    v = v > 0.f ? v : SLOPE * v;              // leaky relu
    alpha[(size_t)e * HH + hd] = v;
    atomicMaxF(&m[d * HH + hd], v);
  }
}

__global__ void edge_exp_sum(const int* __restrict__ ei,
                             const float* __restrict__ m,
                             float* __restrict__ alpha,
                             float* __restrict__ den) {
  int e = blockIdx.x * blockDim.x + threadIdx.x;
  if (e >= ETOT) return;
  int s, d; edge_nodes(ei, e, s, d);
  (void)s;
#pragma unroll
  for (int hd = 0; hd < HH; ++hd) {
    float a = expf(alpha[(size_t)e * HH + hd] - m[d * HH + hd]);
    alpha[(size_t)e * HH + hd] = a;
    atomicAdd(&den[d * HH + hd], a);
  }
}

__global__ void edge_normalize(const int* __restrict__ ei,
                               const float* __restrict__ den,
                               float* __restrict__ alpha) {
  int e = blockIdx.x * blockDim.x + threadIdx.x;
  if (e >= ETOT) return;
  int s, d; edge_nodes(ei, e, s, d);
  (void)s;
#pragma unroll
  for (int hd = 0; hd < HH; ++hd)
    alpha[(size_t)e * HH + hd] /= den[d * HH + hd];
}

// one block per edge, 256 channels
__global__ void edge_aggregate(const int* __restrict__ ei,
                               const float* __restrict__ h,
                               const float* __restrict__ alpha,
                               float* __restrict__ out) {
  int e = blockIdx.x;
  int c = threadIdx.x;
  int s, d; edge_nodes(ei, e, s, d);
  float w = alpha[(size_t)e * HH + (c >> 6)];
  atomicAdd(&out[(size_t)d * FDIM + c], h[(size_t)s * FDIM + c] * w);
}

__global__ void bias_act(float* __restrict__ out, const float* __restrict__ b, int relu) {
  int i = blockIdx.x * blockDim.x + threadIdx.x;
  if (i >= NN * FDIM) return;
  float v = out[i] + b[i & (FDIM - 1)];
  out[i] = (relu && v < 0.f) ? 0.f : v;
}

// ---------------------------------------------------------------- pooling
__global__ void pool_kernel(const float* __restrict__ h, const int* __restrict__ batch,
                            float* gsum, float* gmax, float* cnt) {
  int n = blockIdx.x, c = threadIdx.x;
  int g = batch[n];
  float v = h[(size_t)n * FDIM + c];
  atomicAdd(&gsum[g * FDIM + c], v);
  atomicMaxF(&gmax[g * FDIM + c], v);
  if (c == 0) atomicAdd(&cnt[g], 1.0f);
}

__global__ void pool_finalize(const float* gsum, const float* gmax,
                              const float* cnt, float* g) {
  int gi = blockIdx.x, c = threadIdx.x;   // 512 threads
  g[gi * 512 + c] = (c < FDIM) ? gsum[gi * FDIM + c] / cnt[gi]
                               : gmax[gi * FDIM + (c - FDIM)];
}

// ---------------------------------------------------------------- MLP
__global__ void mlp_layer(const float* __restrict__ in, const float* __restrict__ W,
                          const float* __restrict__ b, float* __restrict__ out,
                          int K, int Cout, int relu) {
  int r = blockIdx.x, c = threadIdx.x;
  if (c >= Cout) return;
  const float* ip = in + (size_t)r * K;
  float s = b[c];
  for (int k = 0; k < K; ++k) s += ip[k] * W[(size_t)k * Cout + c];
  out[(size_t)r * Cout + c] = (relu && s < 0.f) ? 0.f : s;
}

// ---------------------------------------------------------------- launcher
extern "C" void kernel_launch(void* const* d_in, const int* in_sizes, int n_in,
                              void* d_out, int out_size, void* d_ws, size_t ws_size,
                              hipStream_t stream) {
  (void)in_sizes; (void)n_in; (void)out_size; (void)ws_size;
  const float* x    = (const float*)d_in[0];
  const int*   ei   = (const int*)d_in[1];
  const int*   batch= (const int*)d_in[2];
  const float* W1   = (const float*)d_in[3];
  const float* as1  = (const float*)d_in[4];
  const float* ad1  = (const float*)d_in[5];
  const float* b1   = (const float*)d_in[6];
  const float* W2   = (const float*)d_in[7];
  const float* as2  = (const float*)d_in[8];
  const float* ad2  = (const float*)d_in[9];
  const float* b2   = (const float*)d_in[10];
  const float* cw1  = (const float*)d_in[11];
  const float* cb1  = (const float*)d_in[12];
  const float* cw2  = (const float*)d_in[13];
  const float* cb2  = (const float*)d_in[14];
  const float* cw3  = (const float*)d_in[15];
  const float* cb3  = (const float*)d_in[16];
  const float* dw1  = (const float*)d_in[17];
  const float* db1  = (const float*)d_in[18];
  const float* dw2  = (const float*)d_in[19];
  const float* db2  = (const float*)d_in[20];
  const float* dw3  = (const float*)d_in[21];
  const float* db3  = (const float*)d_in[22];
  float* outp = (float*)d_out;

  // workspace carve-up (floats)
  float* ws = (float*)d_ws;
  size_t off = 0;
  float* buf0 = ws + off; off += (size_t)NN * FDIM;   // GEMM output h
  float* buf1 = ws + off; off += (size_t)NN * FDIM;   // aggregated output
  float* es   = ws + off; off += (size_t)NN * HH;
  float* ed   = ws + off; off += (size_t)NN * HH;
  float* m    = ws + off; off += (size_t)NN * HH;
  float* den  = ws + off; off += (size_t)NN * HH;
  float* alp  = ws + off; off += (size_t)ETOT * HH;
  float* gsum = ws + off; off += (size_t)GG * FDIM;
  float* gmax = ws + off; off += (size_t)GG * FDIM;
  float* cnt  = ws + off; off += (size_t)GG;
  float* gcat = ws + off; off += (size_t)GG * 512;
  float* t1   = ws + off; off += (size_t)GG * 256;
  float* t2   = ws + off; off += (size_t)GG * 128;

  const dim3 blk(256);
  const float NEG_INF = -__builtin_huge_valf();
  const int MT   = NN / 16;                       // 1250 M tiles
  const int GRID_GEMM = (MT + 7) / 8;
  const int GRID_NH   = (NN * HH + 255) / 256;
  const int GRID_NF   = (NN * FDIM + 255) / 256;
  const int GRID_E    = (ETOT + 255) / 256;

  // ---------------- layer 1 ----------------
  gemm_wmma_f32<<<GRID_GEMM, blk, 0, stream>>>(x, W1, buf0, NN, 128, FDIM);
  node_scores<<<GRID_NH, blk, 0, stream>>>(buf0, as1, ad1, es, ed);
  fill_f32<<<GRID_NH, blk, 0, stream>>>(m, NEG_INF, NN * HH);
  fill_f32<<<GRID_NH, blk, 0, stream>>>(den, 0.f, NN * HH);
  fill_f32<<<GRID_NF, blk, 0, stream>>>(buf1, 0.f, NN * FDIM);
  edge_alpha_max<<<GRID_E, blk, 0, stream>>>(ei, es, ed, alp, m);
  edge_exp_sum<<<GRID_E, blk, 0, stream>>>(ei, m, alp, den);
  edge_normalize<<<GRID_E, blk, 0, stream>>>(ei, den, alp);
  edge_aggregate<<<dim3(ETOT), blk, 0, stream>>>(ei, buf0, alp, buf1);
  bias_act<<<GRID_NF, blk, 0, stream>>>(buf1, b1, 1);

  // ---------------- layer 2 ----------------
  gemm_wmma_f32<<<GRID_GEMM, blk, 0, stream>>>(buf1, W2, buf0, NN, 256, FDIM);
  node_scores<<<GRID_NH, blk, 0, stream>>>(buf0, as2, ad2, es, ed);
  fill_f32<<<GRID_NH, blk, 0, stream>>>(m, NEG_INF, NN * HH);
  fill_f32<<<GRID_NH, blk, 0, stream>>>(den, 0.f, NN * HH);
  fill_f32<<<GRID_NF, blk, 0, stream>>>(buf1, 0.f, NN * FDIM);
  edge_alpha_max<<<GRID_E, blk, 0, stream>>>(ei, es, ed, alp, m);
  edge_exp_sum<<<GRID_E, blk, 0, stream>>>(ei, m, alp, den);
  edge_normalize<<<GRID_E, blk, 0, stream>>>(ei, den, alp);
  edge_aggregate<<<dim3(ETOT), blk, 0, stream>>>(ei, buf0, alp, buf1);
  bias_act<<<GRID_NF, blk, 0, stream>>>(buf1, b2, 0);   // no relu on layer 2

  // ---------------- pooling ----------------
  fill_f32<<<(GG * FDIM + 255) / 256, blk, 0, stream>>>(gsum, 0.f, GG * FDIM);
  fill_f32<<<(GG * FDIM + 255) / 256, blk, 0, stream>>>(gmax, NEG_INF, GG * FDIM);
  fill_f32<<<1, dim3(GG), 0, stream>>>(cnt, 0.f, GG);
  pool_kernel<<<dim3(NN), blk, 0, stream>>>(buf1, batch, gsum, gmax, cnt);
  pool_finalize<<<dim3(GG), dim3(512), 0, stream>>>(gsum, gmax, cnt, gcat);

  // ---------------- heads ----------------
  mlp_layer<<<dim3(GG), dim3(256), 0, stream>>>(gcat, cw1, cb1, t1, 512, 256, 1);
  mlp_layer<<<dim3(GG), dim3(128), 0, stream>>>(t1, cw2, cb2, t2, 256, 128, 1);
  mlp_layer<<<dim3(GG), dim3(8),   0, stream>>>(t2, cw3, cb3, outp, 128, 8, 0);

  mlp_layer<<<dim3(GG), dim3(256), 0, stream>>>(gcat, dw1, db1, t1, 512, 256, 1);
  mlp_layer<<<dim3(GG), dim3(128), 0, stream>>>(t1, dw2, db2, t2, 256, 128, 1);
  mlp_layer<<<dim3(GG), dim3(8),   0, stream>>>(t2, dw3, db3, outp + GG * 8, 128, 8, 0);
}